// Model_60120952209847
// MI455X (gfx1250) — compile-verified
//
#include <hip/hip_runtime.h>
#include <math.h>

// ---------------------------------------------------------------------------
// Bipartite 3-layer GAT (PyG GATConv semantics) + inner-product decoder.
// MI455X plan:
//  - hs = X @ Ws via V_WMMA_F32_16X16X4_F32 (fp32 matches reference exactly;
//    GEMM volume ~8 GFLOP total, negligible vs matrix-pipe peak).
//  - a_d = x_dst @ (Wd @ ad): dst-side GEMM algebraically collapsed to a GEMV.
//  - Edge phase (dominant, ~3 GB total traffic, ~0.13 ms @ 23.3 TB/s; node
//    features fit in 192MB L2): 3 passes with float atomics, one wave32/edge
//    for the H=64 weighted scatter-add.
// ---------------------------------------------------------------------------

#define H64 64

typedef float v2f __attribute__((ext_vector_type(2)));
typedef float v8f __attribute__((ext_vector_type(8)));

static __device__ __forceinline__ float warp_reduce_sum(float v) {
  #pragma unroll
  for (int off = 16; off > 0; off >>= 1) v += __shfl_down(v, off, 32);
  return v;
}

// Monotonic-bits float atomic max (init with -inf).
static __device__ __forceinline__ void atomicMaxFloat(float* addr, float v) {
  if (v >= 0.0f) atomicMax((int*)addr, __float_as_int(v));
  else           atomicMin((unsigned int*)addr, __float_as_uint(v));
}

// wv[k] = sum_h W[k][h] * a[h]   (K <= 128, H = 64)
__global__ void wvec_kernel(const float* __restrict__ W, const float* __restrict__ a,
                            float* __restrict__ wv, int K) {
  int k = threadIdx.x;
  if (k < K) {
    float s = 0.0f;
    #pragma unroll
    for (int h = 0; h < H64; ++h) s += W[k * H64 + h] * a[h];
    wv[k] = s;
  }
}

// out[row] = dot(X[row, :K], wv)    one wave32 per row
__global__ void gemv_kernel(const float* __restrict__ X, const float* __restrict__ wv,
                            float* __restrict__ out, int M, int K) {
  int row  = blockIdx.x * 8 + (threadIdx.x >> 5);
  int lane = threadIdx.x & 31;
  if (row >= M) return;
  const float* xr = X + (size_t)row * K;
  float s = 0.0f;
  for (int k = lane; k < K; k += 32) s += xr[k] * wv[k];
  s = warp_reduce_sum(s);
  if (lane == 0) out[row] = s;
}

// Hout[M x 64] = X[M x K] @ W[K x 64], K in {64,128}, M % 16 == 0.
// 4 waves / block; each wave computes a 16x64 stripe with 4 f32 WMMA accums.
__global__ void __launch_bounds__(128)
gemm_wmma_kernel(const float* __restrict__ X, const float* __restrict__ W,
                 float* __restrict__ Hout, int M, int K) {
  __shared__ float sW[128 * H64];              // up to 32 KB
  int tid = threadIdx.x;
  for (int i = tid; i < K * H64; i += 128) sW[i] = W[i];
  __syncthreads();

  int wave = tid >> 5;
  int lane = tid & 31;
  int row0 = blockIdx.x * 64 + wave * 16;      // wave-uniform guard -> EXEC all 1s
  if (row0 >= M) return;

  int mrow  = row0 + (lane & 15);
  int khalf = (lane >> 4) << 1;                // lanes 0-15: K k,k+1 ; 16-31: k+2,k+3
  int col   = lane & 15;
  const float* xr = X + (size_t)mrow * K;

  v8f acc0 = {}, acc1 = {}, acc2 = {}, acc3 = {};
  for (int kk = 0; kk < K; kk += 4) {
    v2f a = *(const v2f*)(xr + kk + khalf);    // 8B aligned (kk+khalf even)
    const float* wb = sW + (kk + khalf) * H64;
    v2f b0 = { wb[col +  0], wb[H64 + col +  0] };
    v2f b1 = { wb[col + 16], wb[H64 + col + 16] };
    v2f b2 = { wb[col + 32], wb[H64 + col + 32] };
    v2f b3 = { wb[col + 48], wb[H64 + col + 48] };
    acc0 = __builtin_amdgcn_wmma_f32_16x16x4_f32(false, a, false, b0, (short)0, acc0, false, false);
    acc1 = __builtin_amdgcn_wmma_f32_16x16x4_f32(false, a, false, b1, (short)0, acc1, false, false);
    acc2 = __builtin_amdgcn_wmma_f32_16x16x4_f32(false, a, false, b2, (short)0, acc2, false, false);
    acc3 = __builtin_amdgcn_wmma_f32_16x16x4_f32(false, a, false, b3, (short)0, acc3, false, false);
  }

  // C layout: VGPR v -> rows v (lanes 0-15) / v+8 (lanes 16-31), N = lane&15.
  int rbase = row0 + ((lane >> 4) << 3);
  #pragma unroll
  for (int v = 0; v < 8; ++v) {
    float* o = Hout + (size_t)(rbase + v) * H64 + col;
    o[0] = acc0[v]; o[16] = acc1[v]; o[32] = acc2[v]; o[48] = acc3[v];
  }
}

__global__ void init_m_den_kernel(float* __restrict__ m, float* __restrict__ den, int N) {
  int i = blockIdx.x * blockDim.x + threadIdx.x;
  if (i < N) { m[i] = -INFINITY; den[i] = 0.0f; }
}

// logits[e] = leaky_relu(aS[src]+aD[dst], 0.2); segment max into m[dst]
__global__ void logits_max_kernel(const float* __restrict__ aS, const float* __restrict__ aD,
                                  const int* __restrict__ src, const int* __restrict__ dst,
                                  float* __restrict__ logits, float* __restrict__ m, int E) {
  int e = blockIdx.x * blockDim.x + threadIdx.x;
  if (e >= E) return;
  int d = dst[e];
  float lo = aS[src[e]] + aD[d];
  lo = lo > 0.0f ? lo : 0.2f * lo;
  logits[e] = lo;
  atomicMaxFloat(&m[d], lo);
}

__global__ void fix_m_kernel(float* __restrict__ m, int N) {
  int i = blockIdx.x * blockDim.x + threadIdx.x;
  if (i < N) { float v = m[i]; m[i] = (v == -INFINITY) ? 0.0f : v; }
}

// ex[e] = exp(logits[e]-m[dst]); den[dst] += ex
__global__ void ex_denom_kernel(const float* __restrict__ logits, const float* __restrict__ m,
                                const int* __restrict__ dst, float* __restrict__ ex,
                                float* __restrict__ den, int E) {
  int e = blockIdx.x * blockDim.x + threadIdx.x;
  if (e >= E) return;
  int d = dst[e];
  float v = __expf(logits[e] - m[d]);
  ex[e] = v;
  atomicAdd(&den[d], v);
}

__global__ void init_out_kernel(float* __restrict__ out, const float* __restrict__ b, int Nd) {
  int i = blockIdx.x * blockDim.x + threadIdx.x;
  if (i < Nd * H64) out[i] = b[i & (H64 - 1)];
}

// one wave32 per edge: out[dst,:] += (ex/den[dst]) * hs[src,:]
__global__ void aggregate_kernel(const float* __restrict__ ex, const float* __restrict__ den,
                                 const float* __restrict__ hs,
                                 const int* __restrict__ src, const int* __restrict__ dst,
                                 float* __restrict__ out, int E) {
  int e = blockIdx.x * 8 + (threadIdx.x >> 5);
  int lane = threadIdx.x & 31;
  if (e >= E) return;
  int s = src[e], d = dst[e];
  float alpha = ex[e] / (den[d] + 1e-16f);
  const float* h = hs + (size_t)s * H64;
  float* o = out + (size_t)d * H64;
  atomicAdd(&o[lane],      alpha * h[lane]);
  atomicAdd(&o[lane + 32], alpha * h[lane + 32]);
}

__global__ void relu_kernel(float* __restrict__ x, int n) {
  int i = blockIdx.x * blockDim.x + threadIdx.x;
  if (i < n) x[i] = fmaxf(x[i], 0.0f);
}

// out[l] = dot(xu[ls[l]], xi[ld[l]])  one wave32 per label
__global__ void decoder_kernel(const float* __restrict__ xu, const float* __restrict__ xi,
                               const int* __restrict__ ls, const int* __restrict__ ld,
                               float* __restrict__ out, int L) {
  int l = blockIdx.x * 8 + (threadIdx.x >> 5);
  int lane = threadIdx.x & 31;
  if (l >= L) return;
  const float* a = xu + (size_t)ls[l] * H64;
  const float* b = xi + (size_t)ld[l] * H64;
  float s = a[lane] * b[lane] + a[lane + 32] * b[lane + 32];
  s = warp_reduce_sum(s);
  if (lane == 0) out[l] = s;
}

// ---------------------------------------------------------------------------
struct ConvP { const float *Ws, *Wd, *as_, *ad_, *b; };

static void run_conv(const float* Xs, int Ns, int Ks,
                     const float* Xd, int Nd, int Kd,
                     const ConvP& p, const int* es, const int* ed, int E,
                     float* out,
                     float* HS, float* AS, float* AD, float* M_, float* DEN,
                     float* LOG, float* EX, float* WVS, float* WVD,
                     hipStream_t stream) {
  wvec_kernel<<<1, 128, 0, stream>>>(p.Ws, p.as_, WVS, Ks);
  wvec_kernel<<<1, 128, 0, stream>>>(p.Wd, p.ad_, WVD, Kd);
  gemv_kernel<<<(Ns + 7) / 8, 256, 0, stream>>>(Xs, WVS, AS, Ns, Ks);
  gemv_kernel<<<(Nd + 7) / 8, 256, 0, stream>>>(Xd, WVD, AD, Nd, Kd);
  gemm_wmma_kernel<<<(Ns + 63) / 64, 128, 0, stream>>>(Xs, p.Ws, HS, Ns, Ks);
  init_m_den_kernel<<<(Nd + 255) / 256, 256, 0, stream>>>(M_, DEN, Nd);
  logits_max_kernel<<<(E + 255) / 256, 256, 0, stream>>>(AS, AD, es, ed, LOG, M_, E);
  fix_m_kernel<<<(Nd + 255) / 256, 256, 0, stream>>>(M_, Nd);
  ex_denom_kernel<<<(E + 255) / 256, 256, 0, stream>>>(LOG, M_, ed, EX, DEN, E);
  init_out_kernel<<<(Nd * H64 + 255) / 256, 256, 0, stream>>>(out, p.b, Nd);
  aggregate_kernel<<<(E + 7) / 8, 256, 0, stream>>>(EX, DEN, HS, es, ed, out, E);
}

extern "C" void kernel_launch(void* const* d_in, const int* in_sizes, int n_in,
                              void* d_out, int out_size, void* d_ws, size_t ws_size,
                              hipStream_t stream) {
  (void)n_in; (void)ws_size;
  const float* x_user = (const float*)d_in[0];
  const float* x_item = (const float*)d_in[1];
  const int NU = in_sizes[0] / 128;
  const int NI = in_sizes[1] / 64;

  // params leaves: 6 convs x 5 tensors starting at index 2.
  // Detect flatten order via l1_rates first-leaf size: Ws is 128x64 (insertion
  // order Ws,Wd,as,ad,b) vs Wd 64x64 first (alpha-sorted Wd,Ws,ad,as,b).
  const bool insertion = (in_sizes[2] == 128 * 64);
  auto getConv = [&](int c) {
    int base = 2 + c * 5;
    ConvP p;
    if (insertion) {
      p.Ws  = (const float*)d_in[base + 0];
      p.Wd  = (const float*)d_in[base + 1];
      p.as_ = (const float*)d_in[base + 2];
      p.ad_ = (const float*)d_in[base + 3];
      p.b   = (const float*)d_in[base + 4];
    } else {
      p.Wd  = (const float*)d_in[base + 0];
      p.Ws  = (const float*)d_in[base + 1];
      p.ad_ = (const float*)d_in[base + 2];
      p.as_ = (const float*)d_in[base + 3];
      p.b   = (const float*)d_in[base + 4];
    }
    return p;
  };
  ConvP p1r = getConv(0), p1v = getConv(1), p2r = getConv(2),
        p2v = getConv(3), p3r = getConv(4), p3v = getConv(5);

  const int* rates_src = (const int*)d_in[32];
  const int* rates_dst = (const int*)d_in[33];
  const int* rev_src   = (const int*)d_in[34];
  const int* rev_dst   = (const int*)d_in[35];
  const int* label_src = (const int*)d_in[36];
  const int* label_dst = (const int*)d_in[37];
  const int E = in_sizes[32];
  const int L = in_sizes[36];

  // Workspace layout (floats).
  float* ws = (float*)d_ws;
  size_t off = 0;
  float* U0  = ws + off; off += (size_t)NU * H64;
  float* U1  = ws + off; off += (size_t)NU * H64;
  float* I0  = ws + off; off += (size_t)NI * H64;
  float* I1  = ws + off; off += (size_t)NI * H64;
  float* HS  = ws + off; off += (size_t)NU * H64;   // src-transformed feats (max Ns)
  float* AS  = ws + off; off += (size_t)NU;
  float* AD  = ws + off; off += (size_t)NU;
  float* M_  = ws + off; off += (size_t)NU;
  float* DEN = ws + off; off += (size_t)NU;
  float* LOG = ws + off; off += (size_t)E;
  float* EX  = ws + off; off += (size_t)E;
  float* WVS = ws + off; off += 128;
  float* WVD = ws + off; off += 128;

  // Layer 1 (user D=128 -> item, item D=64 -> user), then ReLU.
  run_conv(x_user, NU, 128, x_item, NI, 64, p1r, rates_src, rates_dst, E, I0,
           HS, AS, AD, M_, DEN, LOG, EX, WVS, WVD, stream);
  run_conv(x_item, NI, 64, x_user, NU, 128, p1v, rev_src, rev_dst, E, U0,
           HS, AS, AD, M_, DEN, LOG, EX, WVS, WVD, stream);
  relu_kernel<<<((size_t)NU * H64 + 255) / 256, 256, 0, stream>>>(U0, NU * H64);
  relu_kernel<<<((size_t)NI * H64 + 255) / 256, 256, 0, stream>>>(I0, NI * H64);

  // Layer 2, then ReLU.
  run_conv(U0, NU, 64, I0, NI, 64, p2r, rates_src, rates_dst, E, I1,
           HS, AS, AD, M_, DEN, LOG, EX, WVS, WVD, stream);
  run_conv(I0, NI, 64, U0, NU, 64, p2v, rev_src, rev_dst, E, U1,
           HS, AS, AD, M_, DEN, LOG, EX, WVS, WVD, stream);
  relu_kernel<<<((size_t)NU * H64 + 255) / 256, 256, 0, stream>>>(U1, NU * H64);
  relu_kernel<<<((size_t)NI * H64 + 255) / 256, 256, 0, stream>>>(I1, NI * H64);

  // Layer 3 (no ReLU).
  run_conv(U1, NU, 64, I1, NI, 64, p3r, rates_src, rates_dst, E, I0,
           HS, AS, AD, M_, DEN, LOG, EX, WVS, WVD, stream);
  run_conv(I1, NI, 64, U1, NU, 64, p3v, rev_src, rev_dst, E, U0,
           HS, AS, AD, M_, DEN, LOG, EX, WVS, WVD, stream);

  // Inner-product decoder.
  decoder_kernel<<<(L + 7) / 8, 256, 0, stream>>>(U0, I0, label_src, label_dst,
                                                  (float*)d_out, L);
  (void)out_size;
}